// MyGINConv_18614388261141
// MI455X (gfx1250) — compile-verified
//
#include <hip/hip_runtime.h>
#include <hip/hip_bf16.h>

// GINConv for MI455X (gfx1250, wave32).
//   agg = x; agg[dst] += x[src] (E atomic f32 adds, L2-resident);
//   out = agg @ W^T + b  via V_WMMA_F32_16X16X4_F32 (exact fp32 path).
//
// Roofline: GEMM is 0.66 GFLOP, all live data (~31 MB) fits the 192 MB L2;
// bound by L2 atomic throughput of the scatter phase -> use hardware
// global_atomic_add_f32 (not a CAS loop) and coalesced 512B/wave gathers.

typedef __attribute__((ext_vector_type(2))) float v2f;
typedef __attribute__((ext_vector_type(4))) float v4f;
typedef __attribute__((ext_vector_type(8))) float v8f;

#define NFEAT 128

// Hardware fp32 atomic add (no-return form, tracked by STOREcnt; s_endpgm's
// implicit wait-idle covers completion). Avoids possible CAS-loop lowering of
// HIP atomicAdd(float*).
__device__ __forceinline__ void gatomic_add_f32(float* p, float v) {
  asm volatile("global_atomic_add_f32 %0, %1, off" : : "v"(p), "v"(v) : "memory");
}

// ---- Kernel 1: agg = x (also folds the "+ x" / eps=0 term) -----------------
__global__ void __launch_bounds__(256) gin_init_kernel(const float* __restrict__ x,
                                                       float* __restrict__ agg,
                                                       int n4) {
  int i = blockIdx.x * blockDim.x + threadIdx.x;
  if (i < n4) ((v4f*)agg)[i] = ((const v4f*)x)[i];
}

// ---- Kernel 2: agg[dst] += x[src], one wave32 per edge ---------------------
// Lane l handles features 4l..4l+3: one coalesced float4 load of the source
// row (512 B per wave, L2-hit after first touch) + 4 hardware f32 atomics.
__global__ void __launch_bounds__(256) gin_scatter_kernel(const float* __restrict__ x,
                                                          const long long* __restrict__ edge_index,
                                                          float* __restrict__ agg,
                                                          int E) {
  int e    = blockIdx.x * (blockDim.x >> 5) + (threadIdx.x >> 5);
  int lane = threadIdx.x & 31;
  if (e >= E) return;
  long long src = edge_index[e];                 // edge_index[0][e]
  long long dst = edge_index[(size_t)E + e];     // edge_index[1][e]
  v4f v = ((const v4f*)(x + (size_t)src * NFEAT))[lane];
  float* d = agg + (size_t)dst * NFEAT + lane * 4;
  gatomic_add_f32(d + 0, v[0]);
  gatomic_add_f32(d + 1, v[1]);
  gatomic_add_f32(d + 2, v[2]);
  gatomic_add_f32(d + 3, v[3]);
}

// ---- Kernel 3: out = agg @ W^T + b  via V_WMMA_F32_16X16X4_F32 -------------
// Block = 256 threads = 8 waves; blockIdx.x picks a 16-row stripe, wave id
// picks one of 8 column tiles. Per wave: 32 WMMAs over K=128 (step 4).
//
// A (16x4 f32, ISA layout): lane L (h=L/16, m=L%16) holds VGPR0/1 =
//   h_mat[row0+m][k0 + 2h + {0,1}]  -> one float2 load.
// B (4x16, mirrored layout): lane L holds B[k0+2h+{0,1}][col0+m] where
//   B[k][n] = W[n][k]  -> float2 load from W row (col0+m).
// C/D (16x16 f32): VGPR i -> row (i + 8h), col m.
__global__ void __launch_bounds__(256) gin_gemm_kernel(const float* __restrict__ h_mat,
                                                       const float* __restrict__ Wm,
                                                       const float* __restrict__ bias,
                                                       float* __restrict__ out) {
  const int lane = threadIdx.x & 31;
  const int wave = threadIdx.x >> 5;
  const int half = lane >> 4;   // 0 or 1
  const int m16  = lane & 15;
  const int row0 = blockIdx.x * 16;
  const int col0 = wave * 16;

  const float* arow = h_mat + (size_t)(row0 + m16) * NFEAT + 2 * half;
  const float* brow = Wm    + (size_t)(col0 + m16) * NFEAT + 2 * half;

  v8f acc = {};
#pragma unroll
  for (int k0 = 0; k0 < NFEAT; k0 += 4) {
    v2f a = *(const v2f*)(arow + k0);
    v2f b = *(const v2f*)(brow + k0);
    // (neg_a, A, neg_b, B, c_mod, C, reuse_a, reuse_b)
    acc = __builtin_amdgcn_wmma_f32_16x16x4_f32(false, a, false, b,
                                                (short)0, acc, false, false);
  }

  const float bv = bias[col0 + m16];
#pragma unroll
  for (int i = 0; i < 8; ++i) {
    out[(size_t)(row0 + i + 8 * half) * NFEAT + col0 + m16] = acc[i] + bv;
  }
}

extern "C" void kernel_launch(void* const* d_in, const int* in_sizes, int n_in,
                              void* d_out, int out_size, void* d_ws, size_t ws_size,
                              hipStream_t stream) {
  const float*     x  = (const float*)d_in[0];       // [20000,128] f32
  const long long* ei = (const long long*)d_in[1];   // [2,640000] i64
  // d_in[2] = edge_weight (unused by GINConv)
  const float*     W  = (const float*)d_in[3];       // [128,128] f32
  const float*     b  = (const float*)d_in[4];       // [128] f32
  float* out = (float*)d_out;
  float* agg = (float*)d_ws;                         // needs 20000*128*4 B

  const int E  = in_sizes[2];            // 640000 edges
  const int N  = in_sizes[0] / NFEAT;    // 20000 nodes
  const int n4 = N * NFEAT / 4;          // float4 count

  gin_init_kernel<<<(n4 + 255) / 256, 256, 0, stream>>>(x, agg, n4);
  gin_scatter_kernel<<<(E + 7) / 8, 256, 0, stream>>>(x, ei, agg, E);
  gin_gemm_kernel<<<N / 16, 256, 0, stream>>>(agg, W, b, out);
}